// LayerNorm_v2_60799557042741
// MI455X (gfx1250) — compile-verified
//
#include <hip/hip_runtime.h>
#include <stdint.h>

// LayerNorm-v2 (running-mean variance) for MI455X / gfx1250.
// Bandwidth-bound: 1 GB traffic @ 23.3 TB/s => ~43us floor. Single pass:
// async global->LDS double-buffered row streaming (ASYNCcnt), wave32 scans,
// registers hold x for the final normalize+store.

#define ROWS 32768
#define COLS 4096
#define TPB  256
#define PER  16          // COLS / TPB
#define NBLK 4096        // 8 rows per block, grid-stride
#define NWAVE (TPB / 32)
#define EPS  1e-5f

#if defined(__HIP_DEVICE_COMPILE__)

#if __has_builtin(__builtin_amdgcn_global_load_async_to_lds_b128)
#define HAVE_ASYNC_BUILTIN 1
#else
#define HAVE_ASYNC_BUILTIN 0
#endif

typedef int v4i __attribute__((vector_size(4 * sizeof(int))));
typedef __attribute__((address_space(1))) v4i gv4i;   // global
typedef __attribute__((address_space(3))) v4i lv4i;   // LDS

__device__ __forceinline__ void async_copy_b128(const float* gsrc, float* ldst) {
#if HAVE_ASYNC_BUILTIN
    __builtin_amdgcn_global_load_async_to_lds_b128(
        (gv4i*)(uintptr_t)gsrc,
        (lv4i*)(uint32_t)(uintptr_t)ldst,
        0, 0);
#else
    // VDST = LDS byte address (low 32 bits of generic ptr), VADDR = 64-bit global addr
    asm volatile("global_load_async_to_lds_b128 %0, %1, off"
                 :: "v"((uint32_t)(uintptr_t)ldst), "v"(gsrc)
                 : "memory");
#endif
}

#if __has_builtin(__builtin_amdgcn_s_wait_asynccnt)
#define WAIT_ASYNC(n) __builtin_amdgcn_s_wait_asynccnt(n)
#else
#define WAIT_ASYNC(n) asm volatile("s_wait_asynccnt %0" :: "n"(n) : "memory")
#endif

#else
// host pass: bodies never emitted
__device__ __forceinline__ void async_copy_b128(const float*, float*) {}
#define WAIT_ASYNC(n) ((void)0)
#endif

__global__ __launch_bounds__(TPB) void lnv2_running_mean_kernel(
    const float* __restrict__ x,
    const float* __restrict__ alpha_p,
    const float* __restrict__ beta_p,
    float* __restrict__ out)
{
    __shared__ float buf[2][COLS];     // 2 x 16 KB double buffer
    __shared__ float wsum[NWAVE];
    __shared__ float wsq[NWAVE];

    const int tid  = threadIdx.x;
    const int lane = tid & 31;
    const int wv   = tid >> 5;

    const float alpha = alpha_p[0];
    const float beta  = beta_p[0];

    // Row-invariant reciprocals 1/(i+1) for running mean — hoisted out of row loop.
    float inv[PER];
#pragma unroll
    for (int k = 0; k < PER; ++k) {
        const int i = tid * PER + k;
#if defined(__HIP_DEVICE_COMPILE__)
        inv[k] = __builtin_amdgcn_rcpf((float)(i + 1));
#else
        inv[k] = 1.0f / (float)(i + 1);
#endif
    }

    const int row0    = blockIdx.x;
    const int rstride = gridDim.x;

    // Prefetch first row into buffer 0: 4 x b128 per thread (16 KB / 256 threads).
    {
        const float* src = x + (size_t)row0 * COLS;
#pragma unroll
        for (int j = 0; j < 4; ++j) {
            const int off = j * 1024 + tid * 4;   // float index, coalesced per wave
            async_copy_b128(src + off, &buf[0][off]);
        }
    }

    int parity = 0;
    for (int row = row0; row < ROWS; row += rstride, parity ^= 1) {
        const int  nrow = row + rstride;
        const bool pref = (nrow < ROWS);

        if (pref) {
            // Prefetch next row into the other buffer, then wait only for the
            // current row's 4 async ops (<=4 outstanding keeps prefetch in flight).
            const float* src = x + (size_t)nrow * COLS;
#pragma unroll
            for (int j = 0; j < 4; ++j) {
                const int off = j * 1024 + tid * 4;
                async_copy_b128(src + off, &buf[parity ^ 1][off]);
            }
            WAIT_ASYNC(4);
        } else {
            WAIT_ASYNC(0);
        }
        __syncthreads();   // all waves' async writes for buf[parity] visible

        // Each thread owns 16 contiguous elements of the row.
        float xv[PER];
        {
            const float4* p = (const float4*)&buf[parity][tid * PER];
#pragma unroll
            for (int j = 0; j < 4; ++j) {
                const float4 v = p[j];
                xv[4 * j + 0] = v.x; xv[4 * j + 1] = v.y;
                xv[4 * j + 2] = v.z; xv[4 * j + 3] = v.w;
            }
        }

        // Thread-local total.
        float tot = 0.f;
#pragma unroll
        for (int k = 0; k < PER; ++k) tot += xv[k];

        // wave32 inclusive scan of per-thread totals.
        float sc = tot;
#pragma unroll
        for (int d = 1; d < 32; d <<= 1) {
            const float o = __shfl_up(sc, d, 32);
            if (lane >= d) sc += o;
        }
        if (lane == 31) wsum[wv] = sc;   // wave total
        __syncthreads();

        float waveExcl = 0.f, rowTot = 0.f;
#pragma unroll
        for (int w = 0; w < NWAVE; ++w) {
            const float ws = wsum[w];
            if (w < wv) waveExcl += ws;
            rowTot += ws;
        }
        // Exclusive prefix of everything before this thread's 16-element segment.
        const float base = waveExcl + (sc - tot);

        // Squared deviation from the running (cumulative) mean.
        float c = 0.f, sq = 0.f;
#pragma unroll
        for (int k = 0; k < PER; ++k) {
            c += xv[k];
            const float rm = (base + c) * inv[k];   // cumsum_i / (i+1)
            const float d2 = xv[k] - rm;
            sq = fmaf(d2, d2, sq);
        }

        // wave32 reduce, then 8-wave combine.
#pragma unroll
        for (int d = 16; d > 0; d >>= 1) sq += __shfl_xor(sq, d, 32);
        if (lane == 0) wsq[wv] = sq;
        __syncthreads();
        float sqTot = 0.f;
#pragma unroll
        for (int w = 0; w < NWAVE; ++w) sqTot += wsq[w];

        const float mean = rowTot * (1.0f / (float)COLS);
        const float var  = sqTot * (1.0f / (float)(COLS - 1));
#if defined(__HIP_DEVICE_COMPILE__)
        const float rstd = __builtin_amdgcn_rsqf(var + EPS);
#else
        const float rstd = rsqrtf(var + EPS);
#endif
        const float scale = alpha * rstd;

        // Normalize from registers, b128 stores.
        float4* op = (float4*)(out + (size_t)row * COLS + tid * PER);
#pragma unroll
        for (int j = 0; j < 4; ++j) {
            float4 v;
            v.x = (xv[4 * j + 0] - mean) * scale + beta;
            v.y = (xv[4 * j + 1] - mean) * scale + beta;
            v.z = (xv[4 * j + 2] - mean) * scale + beta;
            v.w = (xv[4 * j + 3] - mean) * scale + beta;
            op[j] = v;
        }
        __syncthreads();   // buf[parity] and wsum/wsq may be rewritten next iter
    }
}

extern "C" void kernel_launch(void* const* d_in, const int* in_sizes, int n_in,
                              void* d_out, int out_size, void* d_ws, size_t ws_size,
                              hipStream_t stream) {
    (void)in_sizes; (void)n_in; (void)out_size; (void)d_ws; (void)ws_size;
    const float* xp    = (const float*)d_in[0];
    const float* alpha = (const float*)d_in[1];
    const float* beta  = (const float*)d_in[2];
    float* op          = (float*)d_out;
    lnv2_running_mean_kernel<<<NBLK, TPB, 0, stream>>>(xp, alpha, beta, op);
}